// SparseMultiheadAttention_14628658610667
// MI455X (gfx1250) — compile-verified
//
#include <hip/hip_runtime.h>
#include <hip/hip_bf16.h>

typedef float v2f __attribute__((ext_vector_type(2)));
typedef float v8f __attribute__((ext_vector_type(8)));

#define EMBED 128
#define NHEAD 8
#define HDIM 16

// ---------------------------------------------------------------------------
// Generic projection: out[t, n] = (sum_k in[t,k] * W[n,k] + bias[n]) * scale
// One wave computes a 16-row stripe across all 128 output columns using
// V_WMMA_F32_16X16X4_F32 (fp32 A/B/C, K=4 per step, 32 steps over K=128).
// A-fragment layout (ISA 7.12.2, 32-bit A 16x4): lane L, vgpr j ->
//   A[M = L%16][K = 4*kk + 2*(L/16) + j]  -> one float2 load per k-step.
// B = W^T, so B[k][n] = W[n][k]; lane L, vgpr j ->
//   W[n = ct*16 + L%16][k = 4*kk + 2*(L/16) + j] -> same float2 pattern.
// D layout: vgpr v, lane L -> row (v + 8*(L/16)), col (ct*16 + L%16).
// ---------------------------------------------------------------------------
__global__ void proj_wmma_f32(const float* __restrict__ in,
                              const float* __restrict__ W,
                              const float* __restrict__ bias,
                              float scale,
                              float* __restrict__ out,
                              int nTiles) {
  const int wave = (blockIdx.x * blockDim.x + threadIdx.x) >> 5;  // wave32
  if (wave >= nTiles) return;          // wave-uniform: EXEC stays all-ones
  const int lane = threadIdx.x & 31;
  const int row = lane & 15;
  const int hi  = lane >> 4;           // 0 or 1

  const int tbase = wave * 16;
  const float* arow = in + (tbase + row) * EMBED + 2 * hi;

  // Load the full A stripe for this wave once; reused by all 8 column tiles.
  v2f a_frag[32];
#pragma unroll
  for (int kk = 0; kk < 32; ++kk)
    a_frag[kk] = *reinterpret_cast<const v2f*>(arow + 4 * kk);

  for (int ct = 0; ct < 8; ++ct) {
    const float* brow = W + (ct * 16 + row) * EMBED + 2 * hi;
    v8f c = {};
#pragma unroll
    for (int kk = 0; kk < 32; ++kk) {
      v2f b = *reinterpret_cast<const v2f*>(brow + 4 * kk);
      c = __builtin_amdgcn_wmma_f32_16x16x4_f32(
          /*neg_a=*/false, a_frag[kk], /*neg_b=*/false, b,
          /*c_mod=*/(short)0, c, /*reuse_a=*/false, /*reuse_b=*/false);
    }
    const int col = ct * 16 + row;
    const float bc = bias ? bias[col] : 0.0f;
#pragma unroll
    for (int v = 0; v < 8; ++v)
      out[(tbase + v + 8 * hi) * EMBED + col] = (c[v] + bc) * scale;
  }
}

// ---------------------------------------------------------------------------
// Edge phase. One thread per (edge, head); 8 consecutive lanes = 8 heads of
// one edge (blockDim multiple of 8). Segment max/sum via L2 atomics.
// ---------------------------------------------------------------------------
__device__ __forceinline__ unsigned float_to_ordered(float f) {
  unsigned u = __float_as_uint(f);
  return (u & 0x80000000u) ? ~u : (u | 0x80000000u);
}
__device__ __forceinline__ float ordered_to_float(unsigned u) {
  return (u & 0x80000000u) ? __uint_as_float(u ^ 0x80000000u)
                           : __uint_as_float(~u);
}

__global__ void edge_scores(const float* __restrict__ Q,
                            const float* __restrict__ K,
                            const int* __restrict__ edge,
                            float* __restrict__ scores,
                            unsigned* __restrict__ nmax, int nEdges) {
  const int gid = blockIdx.x * blockDim.x + threadIdx.x;
  if (gid >= nEdges * NHEAD) return;
  const int e = gid >> 3, h = gid & 7;
  const int dst = edge[e];
  const int src = edge[nEdges + e];
  const float4* qp = reinterpret_cast<const float4*>(Q + dst * EMBED + h * HDIM);
  const float4* kp = reinterpret_cast<const float4*>(K + src * EMBED + h * HDIM);
  float s = 0.0f;
#pragma unroll
  for (int i = 0; i < 4; ++i) {
    float4 a = qp[i], b = kp[i];
    s += a.x * b.x + a.y * b.y + a.z * b.z + a.w * b.w;
  }
  scores[gid] = s;
  atomicMax(&nmax[dst * NHEAD + h], float_to_ordered(s));
}

__global__ void edge_exp(float* __restrict__ sc,           // scores -> ex in place
                         const int* __restrict__ edge,
                         const unsigned* __restrict__ nmax,
                         float* __restrict__ denom, int nEdges) {
  const int gid = blockIdx.x * blockDim.x + threadIdx.x;
  if (gid >= nEdges * NHEAD) return;
  const int e = gid >> 3, h = gid & 7;
  const int dst = edge[e];
  const float m = ordered_to_float(nmax[dst * NHEAD + h]);
  const float x = __expf(sc[gid] - m);
  sc[gid] = x;
  atomicAdd(&denom[dst * NHEAD + h], x);
}

__global__ void edge_scatter(const float* __restrict__ ex,
                             const int* __restrict__ edge,
                             const float* __restrict__ denom,
                             const float* __restrict__ V,
                             float* __restrict__ accum,       // [T,128], zeroed
                             float* __restrict__ attn_mean,   // [E]
                             int nEdges) {
  const int gid = blockIdx.x * blockDim.x + threadIdx.x;
  if (gid >= nEdges * NHEAD) return;
  const int e = gid >> 3, h = gid & 7;
  const int dst = edge[e];
  const int src = edge[nEdges + e];
  const float a = ex[gid] / (denom[dst * NHEAD + h] + 1e-16f);

  // mean over the 8 heads of this edge (8 consecutive lanes, wave32)
  float msum = a;
  msum += __shfl_xor(msum, 1, 8);
  msum += __shfl_xor(msum, 2, 8);
  msum += __shfl_xor(msum, 4, 8);
  if (h == 0) attn_mean[e] = msum * 0.125f;

  const float* vp = V + src * EMBED + h * HDIM;
  float* op = accum + dst * EMBED + h * HDIM;
#pragma unroll
  for (int d = 0; d < HDIM; ++d)
    atomicAdd(&op[d], a * vp[d]);
}

__global__ void fill_zero(float* __restrict__ p, long n) {
  long i = (long)blockIdx.x * blockDim.x + threadIdx.x;
  if (i < n) p[i] = 0.0f;
}

// ---------------------------------------------------------------------------
extern "C" void kernel_launch(void* const* d_in, const int* in_sizes, int n_in,
                              void* d_out, int out_size, void* d_ws, size_t ws_size,
                              hipStream_t stream) {
  const float* query = (const float*)d_in[0];
  const float* key   = (const float*)d_in[1];
  const float* value = (const float*)d_in[2];
  const int*   edge  = (const int*)d_in[3];
  const float* Wq = (const float*)d_in[4];
  const float* bq = (const float*)d_in[5];
  const float* Wk = (const float*)d_in[6];
  const float* Wv = (const float*)d_in[7];
  const float* Wo = (const float*)d_in[8];
  const float* bo = (const float*)d_in[9];

  const int nNodes = in_sizes[0] / EMBED;       // 50000
  const int nEdges = in_sizes[3] / 2;           // 1600000
  const int nTiles = nNodes / 16;               // 3125 (exact)

  float* out = (float*)d_out;                          // [nNodes,128], also accum
  float* attn_mean = out + (long)nNodes * EMBED;       // [nEdges]

  float* Q  = (float*)d_ws;
  float* K  = Q + (long)nNodes * EMBED;
  float* V  = K + (long)nNodes * EMBED;
  float* SC = V + (long)nNodes * EMBED;                // [nEdges*8], scores->ex
  unsigned* NMAX = (unsigned*)(SC + (long)nEdges * NHEAD);
  float* DEN = (float*)(NMAX + (long)nNodes * NHEAD);  // contiguous with NMAX

  const int B = 256;

  // Zero the attention accumulator (out region) and NMAX+DEN (0 is below every
  // ordered-float encoding, so it's a valid -inf for the max).
  {
    long n1 = (long)nNodes * EMBED;
    fill_zero<<<(int)((n1 + B - 1) / B), B, 0, stream>>>(out, n1);
    long n2 = (long)nNodes * NHEAD * 2;                // NMAX then DEN
    fill_zero<<<(int)((n2 + B - 1) / B), B, 0, stream>>>((float*)NMAX, n2);
  }

  // Projections: q = (x Wq^T + bq) * 1/sqrt(16);  k = x Wk^T;  v = x Wv^T
  const int projGrid = (nTiles + (B / 32) - 1) / (B / 32);
  proj_wmma_f32<<<projGrid, B, 0, stream>>>(query, Wq, bq, 0.25f, Q, nTiles);
  proj_wmma_f32<<<projGrid, B, 0, stream>>>(key,   Wk, nullptr, 1.0f, K, nTiles);
  proj_wmma_f32<<<projGrid, B, 0, stream>>>(value, Wv, nullptr, 1.0f, V, nTiles);

  // Edge softmax + scatter
  const long nWork = (long)nEdges * NHEAD;
  const int eGrid = (int)((nWork + B - 1) / B);
  edge_scores <<<eGrid, B, 0, stream>>>(Q, K, edge, SC, NMAX, nEdges);
  edge_exp    <<<eGrid, B, 0, stream>>>(SC, edge, NMAX, DEN, nEdges);
  edge_scatter<<<eGrid, B, 0, stream>>>(SC, edge, DEN, V, out, attn_mean, nEdges);

  // Output projection in place: out = accum Wo^T + bo.
  // Safe: each wave loads its full 16-row A stripe into registers before any store,
  // and row tiles are disjoint across waves.
  proj_wmma_f32<<<projGrid, B, 0, stream>>>(out, Wo, bo, 1.0f, out, nTiles);
}